// ScatterNL_78022375899653
// MI455X (gfx1250) — compile-verified
//
#include <hip/hip_runtime.h>

typedef float v4f __attribute__((ext_vector_type(4)));

#define C_DIM   3
#define H_DIM   512
#define W_DIM   512
#define PS      7
#define PT      2
#define PATCH_N (PT * C_DIM * PS * PS)   // 294 floats per patch

__device__ __forceinline__ unsigned out_index(int t, int h, int w, unsigned r) {
  // r in [0, 294): decompose into (dt, c, di, dj); all divisors are
  // compile-time constants -> mulhi/shift, no real divides.
  unsigned dt = (r >= 147u) ? 1u : 0u;
  unsigned r2 = r - dt * 147u;          // 0..146
  unsigned c  = r2 / 49u;               // 0..2
  unsigned r3 = r2 - c * 49u;           // 0..48
  unsigned di = r3 / 7u;                // 0..6
  unsigned dj = r3 - di * 7u;           // 0..6
  unsigned tt = (unsigned)t + dt;
  unsigned hh = (unsigned)h + di;
  unsigned ww = (unsigned)w + dj;
  // layout [T, C, H, W]
  return ((tt * C_DIM + c) * H_DIM + hh) * W_DIM + ww;
}

// Hot kernel: straight-line fast path. One thread = 4 consecutive patch
// elements via one non-temporal b128 load, then 4 no-return fp32 atomics.
// Exactly ONE division by 294 per thread; the patch-boundary test is derived
// from the remainder (r0 + 3 >= 294), not a second division.
__global__ __launch_bounds__(256) void ScatterNL_scatter_kernel(
    const float* __restrict__ patches,
    const int*   __restrict__ qinds,
    float*       __restrict__ out,
    unsigned n4full /* = n/4 : number of full float4 groups */) {
  unsigned i = blockIdx.x * 256u + threadIdx.x;
  if (i >= n4full) return;
  unsigned e0 = i * 4u;

  v4f v = __builtin_nontemporal_load((const v4f*)(patches + e0));

  unsigned q0 = e0 / (unsigned)PATCH_N;                 // single magic-div
  unsigned r0 = e0 - q0 * (unsigned)PATCH_N;            // 0..293
  unsigned split_k = (unsigned)PATCH_N - r0;            // k >= split_k -> next patch

  int t0 = qinds[3u * q0 + 0u];   // compiler combines into global_load_b96
  int h0 = qinds[3u * q0 + 1u];
  int w0 = qinds[3u * q0 + 2u];
  int tL = t0, hL = h0, wL = w0;
  if (split_k <= 3u) {            // boundary crossed inside this float4 (~0.7%)
    unsigned qL = q0 + 1u;
    tL = qinds[3u * qL + 0u];
    hL = qinds[3u * qL + 1u];
    wL = qinds[3u * qL + 2u];
  }

#pragma unroll
  for (unsigned k = 0; k < 4u; ++k) {
    bool     second = (k >= split_k);   // false whenever no crossing occurs
    int      t      = second ? tL : t0;
    int      h      = second ? hL : h0;
    int      w      = second ? wL : w0;
    unsigned r      = second ? (k - split_k) : (r0 + k);
    unsigned oidx   = out_index(t, h, w, r);
    // relaxed / agent scope / result discarded -> global_atomic_add_f32 (no rtn)
    __hip_atomic_fetch_add(out + oidx, v[k], __ATOMIC_RELAXED,
                           __HIP_MEMORY_SCOPE_AGENT);
  }
}

// Tail kernel: handles the last n%4 elements (never launched for the
// reference shapes; kept for generality). Single block.
__global__ __launch_bounds__(32) void ScatterNL_tail_kernel(
    const float* __restrict__ patches,
    const int*   __restrict__ qinds,
    float*       __restrict__ out,
    unsigned e_begin, unsigned n) {
  unsigned e = e_begin + threadIdx.x;
  if (e >= n) return;
  float    val = patches[e];
  unsigned q   = e / (unsigned)PATCH_N;
  unsigned r   = e - q * (unsigned)PATCH_N;
  int t = qinds[3u * q + 0u];
  int h = qinds[3u * q + 1u];
  int w = qinds[3u * q + 2u];
  __hip_atomic_fetch_add(out + out_index(t, h, w, r), val,
                         __ATOMIC_RELAXED, __HIP_MEMORY_SCOPE_AGENT);
}

// Copy the accumulation base (vid2fill) into d_out. NT hint on the read-once
// base stream; regular stores install the output working set into L2 ahead of
// the atomic storm.
__global__ __launch_bounds__(256) void ScatterNL_init_kernel(
    const float* __restrict__ base, float* __restrict__ out, int n) {
  int i4 = (blockIdx.x * 256 + threadIdx.x) * 4;
  if (i4 + 3 < n) {
    v4f v = __builtin_nontemporal_load((const v4f*)(base + i4));
    *(v4f*)(out + i4) = v;
  } else {
    for (int k = i4; k < n; ++k) out[k] = base[k];
  }
}

extern "C" void kernel_launch(void* const* d_in, const int* in_sizes, int n_in,
                              void* d_out, int out_size, void* d_ws, size_t ws_size,
                              hipStream_t stream) {
  const float* vid2fill = (const float*)d_in[0];  // [T,C,H,W] accumulation base
  const float* patches  = (const float*)d_in[1];  // [Q,PT,C,PS,PS]
  const int*   qinds    = (const int*)d_in[2];    // [Q,3]
  float*       out      = (float*)d_out;

  // 1) Initialize output with the accumulation base (harness poisons d_out).
  int init_threads = (out_size + 3) / 4;
  int init_blocks  = (init_threads + 255) / 256;
  ScatterNL_init_kernel<<<init_blocks, 256, 0, stream>>>(vid2fill, out, out_size);

  // 2) Scatter-add all full float4 groups (same stream -> ordered after init).
  unsigned n      = (unsigned)in_sizes[1];        // Q * 294
  unsigned n4full = n / 4u;
  unsigned blocks = (n4full + 255u) / 256u;
  if (blocks)
    ScatterNL_scatter_kernel<<<blocks, 256, 0, stream>>>(patches, qinds, out,
                                                         n4full);

  // 3) Remainder (n % 4 elements) — pure function of input sizes, so the
  //    captured graph is identical across calls.
  unsigned rem = n - n4full * 4u;
  if (rem)
    ScatterNL_tail_kernel<<<1, 32, 0, stream>>>(patches, qinds, out,
                                                n4full * 4u, n);
}